// MemoryWriter_23845658428023
// MI455X (gfx1250) — compile-verified
//
#include <hip/hip_runtime.h>
#include <stdint.h>

// Problem geometry (from the reference)
#define NUM_MEM_ROWS 2048
#define WORD_SIZE    512
#define BATCH        64

// 16-byte (float4) granularity
#define V4_PER_ROW       (WORD_SIZE / 4)                 // 128
#define V4_PER_BATCH     (NUM_MEM_ROWS * V4_PER_ROW)     // 262144
#define V4_TOTAL         (BATCH * V4_PER_BATCH)          // 16777216
#define THREADS          256
#define V4_PER_THREAD    8
#define V4_PER_BLOCK     (THREADS * V4_PER_THREAD)       // 2048 (= 32 KB LDS)
#define NBLOCKS          (V4_TOTAL / V4_PER_BLOCK)       // 8192
#define BLOCKS_PER_BATCH (V4_PER_BATCH / V4_PER_BLOCK)   // 128 -> blocks never straddle batches

__global__ __launch_bounds__(THREADS)
void memwriter_async_copy(const float* __restrict__ z,
                          const float* __restrict__ mem,
                          const int*   __restrict__ wc,
                          float* __restrict__ out)
{
    __shared__ __align__(16) unsigned char smem[V4_PER_BLOCK * 16];

    const uint32_t tid = threadIdx.x;
    const uint32_t blk = blockIdx.x;
    const uint32_t b   = blk >> 7;                 // batch index (128 blocks per batch)
    const uint32_t widx = (uint32_t)(wc[b]) & (NUM_MEM_ROWS - 1); // counter >= 0 in reference

    // Low 32 bits of a flat shared pointer == LDS byte offset of this workgroup's alloc.
    const uint32_t ldsBase = ((uint32_t)(uintptr_t)(&smem[0])) + tid * 16u;
    const uint64_t srcBase = (uint64_t)(uintptr_t)mem;
    const uint64_t dstBase = (uint64_t)(uintptr_t)out;
    const uint64_t zBase   = (uint64_t)(uintptr_t)z + (uint64_t)b * (WORD_SIZE * 4u);

    const uint32_t f0 = blk * (uint32_t)V4_PER_BLOCK + tid;

    // Stage 1: 8 async 16B transfers global -> LDS per lane (ASYNCcnt-tracked, VGPR-data-free).
    // Branchless source-address select so all 8 loads issue back-to-back (max MLP per wave).
#pragma unroll
    for (int i = 0; i < V4_PER_THREAD; ++i) {
        const uint32_t f          = f0 + (uint32_t)i * THREADS;
        const uint32_t rowInBatch = (f >> 7) & (NUM_MEM_ROWS - 1);
        const uint32_t col        = f & (V4_PER_ROW - 1);

        const uint64_t memAddr = srcBase + (uint64_t)f * 16u;
        const uint64_t zAddr   = zBase + (uint64_t)col * 16u;
        const uint64_t m       = 0ull - (uint64_t)(rowInBatch == widx); // all-ones if written row
        const uint64_t src     = (memAddr & ~m) | (zAddr & m);

        const uint32_t lds = ldsBase + (uint32_t)i * (THREADS * 16u);
        asm volatile("global_load_async_to_lds_b128 %0, %1, off th:TH_LOAD_NT"
                     :: "v"(lds), "v"(src) : "memory");
    }

    // Wait for this wave's async loads to land in LDS.
    asm volatile("s_wait_asynccnt 0x0" ::: "memory");

    // Stage 2: 8 async 16B transfers LDS -> global per lane (streamed, non-temporal).
#pragma unroll
    for (int i = 0; i < V4_PER_THREAD; ++i) {
        const uint32_t f   = f0 + (uint32_t)i * THREADS;
        const uint64_t dst = dstBase + (uint64_t)f * 16u;
        const uint32_t lds = ldsBase + (uint32_t)i * (THREADS * 16u);
        asm volatile("global_store_async_from_lds_b128 %0, %1, off th:TH_STORE_NT"
                     :: "v"(dst), "v"(lds) : "memory");
    }

    // new_state = write_counter + 1, appended after the 67,108,864 floats of new_mem.
    if (blk == 0 && tid < BATCH) {
        out[(size_t)V4_TOTAL * 4 + tid] = (float)(wc[tid] + 1);
    }
    // Async stores drain at s_endpgm (implicit wait-idle).
}

extern "C" void kernel_launch(void* const* d_in, const int* in_sizes, int n_in,
                              void* d_out, int out_size, void* d_ws, size_t ws_size,
                              hipStream_t stream)
{
    (void)in_sizes; (void)n_in; (void)out_size; (void)d_ws; (void)ws_size;
    const float* z   = (const float*)d_in[0];
    const float* mem = (const float*)d_in[1];
    const int*   wc  = (const int*)d_in[2];
    float*       out = (float*)d_out;

    memwriter_async_copy<<<NBLOCKS, THREADS, 0, stream>>>(z, mem, wc, out);
}